// BinaryBoundarySoftDice_89335319757443
// MI455X (gfx1250) — compile-verified
//
#include <hip/hip_runtime.h>
#include <math.h>

// BinaryBoundarySoftDice on MI455X (gfx1250), compile-only target.
// dist = (1 + min(d_inf_to_edge, 21)) / 22 ; weight = 2/(1+exp(10*dist)) -> 22-entry LUT.
// Memory roofline: outputs(33.5MB) + masks(33.5MB) ~ 67MB -> ~3us @ 23.3TB/s.
// CDNA5 paths: async global->LDS mask staging (global_load_async_to_lds_b8 + s_wait_asynccnt).
// Binary stages are bit-sliced (ballot-packed u32 words) to minimize LDS-issue pressure.

#define HH 256
#define WW 256
#define NSLICE 128     // B*D = 2*64
#define BAND 32
#define NBAND 8        // HH/BAND
#define RAD 21         // distance cap: acc = max(0, 21 - d_inf)
#define RM 76          // BAND + 2*RAD + 2  (mask rows incl. edge halo)
#define RE 74          // BAND + 2*RAD      (edge rows)
#define SGW 304        // G LDS row stride (256 + 2*22 pad, data at col x+22)

__device__ __forceinline__ int imin(int a, int b) { return a < b ? a : b; }
__device__ __forceinline__ int imax(int a, int b) { return a > b ? a : b; }

__device__ __forceinline__ unsigned wave_ballot32(int pred)
{
#if __has_builtin(__builtin_amdgcn_ballot_w32)
    return __builtin_amdgcn_ballot_w32(pred != 0);
#else
    return (unsigned)__ballot(pred);
#endif
}

__global__ __launch_bounds__(256)
void bbsd_main(const float* __restrict__ outp, const int* __restrict__ maskp,
               double* __restrict__ acc6, double* __restrict__ part, int use_part)
{
    __shared__ uint8_t  sm[RM][WW];     // raw mask bytes (async-load destination)
    __shared__ unsigned mb[RM][10];     // bit-packed mask rows; data at k=1..8, k=0/9 zero pads
    __shared__ unsigned eb[RE][8];      // bit-packed edge rows
    __shared__ uint8_t  sg[BAND][SGW];  // vertical capped distance, col offset +22, pads=21
    __shared__ float    wtab[22];       // weight(D) = 2/(1+exp(10*(1+D)/22)), D in [0,21]
    __shared__ double   red[3][8];

    const int x     = threadIdx.x;      // 0..255 == column
    const int lane  = x & 31;
    const int kw    = x >> 5;           // wave id == 32-column group
    const int slice = blockIdx.y;       // 0..127
    const int band  = blockIdx.x;       // 0..7
    const int y0    = band * BAND;
    const int bid   = slice * NBAND + band;

    if (x < 22)
        wtab[x] = 2.0f / (1.0f + __expf(10.0f * (float)(1 + x) / 22.0f));
    if (x < RM) { mb[x][0] = 0u; mb[x][9] = 0u; }   // word pads -> zero image border

    // ---- stage 1: mask band -> LDS bytes via CDNA5 async global->LDS (LSB of int32) ----
    const uint8_t* gbase = (const uint8_t*)maskp + (size_t)slice * HH * WW * 4u;
    for (int mr = 0; mr < RM; ++mr) {
        const int y = y0 - (RAD + 1) + mr;
        if ((unsigned)y < (unsigned)HH) {
            const uint8_t* gp = gbase + ((size_t)y * WW + x) * 4u;  // little-endian LSB
            unsigned ldsa = (unsigned)(size_t)&sm[mr][x];           // low 32 bits = LDS offset
            asm volatile("global_load_async_to_lds_b8 %0, %1, off"
                         :: "v"(ldsa), "v"(gp) : "memory");
        } else {
            sm[mr][x] = 0;              // zero padding outside the image
        }
    }
    asm volatile("s_wait_asynccnt 0" ::: "memory");
    __syncthreads();

    // ---- stage 1.5: bit-pack mask rows: one ballot per row per wave ----
    for (int r = 0; r < RM; ++r) {
        const unsigned bal = wave_ballot32((int)sm[r][x]);
        if (lane == 0) mb[r][kw + 1] = bal;
    }
    __syncthreads();

    // ---- stage 2: edge bits = m & ~(up & down & west & east)  (Laplacian>0, zero pad) ----
    for (int w = x; w < RE * 8; w += 256) {
        const int r  = w >> 3;
        const int kc = (w & 7) + 1;
        const unsigned m  = mb[r + 1][kc];
        const unsigned up = mb[r][kc];
        const unsigned dn = mb[r + 2][kc];
        const unsigned wl = (m << 1) | (mb[r + 1][kc - 1] >> 31);
        const unsigned wr = (m >> 1) | (mb[r + 1][kc + 1] << 31);
        eb[r][kc - 1] = m & ~(up & dn & wl & wr);
    }
    __syncthreads();

    // ---- stage 3: vertical capped L_inf distance via sliding 43-bit window per column.
    //      eb[r][kw] is wave-uniform -> LDS broadcast loads, then per-lane bit extract. ----
    {
        unsigned long long win = 0ull;
        for (int r = 0; r < 2 * RAD + 1; ++r)
            win |= ((unsigned long long)((eb[r][kw] >> lane) & 1u)) << r;
        for (int oy = 0; oy < BAND; ++oy) {
            // bit i of win <-> edge row oy+i, delta = i - 21
            const unsigned up = (unsigned)(win >> RAD);        // delta >= 0 (22 bits)
            const unsigned dn = (unsigned)win & 0x3FFFFFu;     // delta <= 0 (bits 0..21)
            const int gu = up ? __builtin_ctz(up) : RAD;
            const int gd = dn ? (RAD - (31 - __builtin_clz(dn))) : RAD;
            const int g  = imin(imin(gu, gd), RAD);
            sg[oy][x + 22] = (uint8_t)g;
            if (x < 22) { sg[oy][x] = RAD; sg[oy][x + 278] = RAD; }  // horiz pads (=cap)
            if (oy < BAND - 1)
                win = (win >> 1) |
                      (((unsigned long long)((eb[oy + 2 * RAD + 2][kw] >> lane) & 1u)) << (2 * RAD + 1));
        }
    }
    __syncthreads();

    // ---- stage 4: horizontal min-max cone (early exit) + weight LUT + fused dice sums ----
    double aI = 0.0, aA = 0.0, aT = 0.0;
    const float* obase = outp + ((size_t)slice * HH + y0) * WW;
    for (int i = x; i < BAND * WW; i += 256) {
        const int oy = i >> 8;
        const int xx = i & 255;          // == x -> coalesced global reads
        int Dv = sg[oy][xx + 22];
        for (int d = 1; d <= RAD; ++d) {
            if (Dv <= d) break;          // remaining terms are >= d >= Dv
            const int t = imin(sg[oy][xx + 22 - d], sg[oy][xx + 22 + d]);
            Dv = imin(Dv, imax(d, t));
        }
        const float w  = wtab[Dv];
        const float o  = obase[(size_t)oy * WW + xx];
        const float m  = (float)((mb[oy + 22][kw + 1] >> lane) & 1u);  // broadcast word
        const float ow = o * w;
        aA += (double)ow;
        aI += (double)(ow * m);
        aT += (double)(w * m);
    }

    // ---- block reduction ----
    for (int off = 16; off; off >>= 1) {
        aI += __shfl_down(aI, off);
        aA += __shfl_down(aA, off);
        aT += __shfl_down(aT, off);
    }
    if (lane == 0) { red[0][kw] = aI; red[1][kw] = aA; red[2][kw] = aT; }
    __syncthreads();
    if (x == 0) {
        double sI = 0.0, sA = 0.0, sT = 0.0;
        for (int wi = 0; wi < 8; ++wi) { sI += red[0][wi]; sA += red[1][wi]; sT += red[2][wi]; }
        if (use_part) {
            part[(size_t)bid * 3 + 0] = sI;
            part[(size_t)bid * 3 + 1] = sA;
            part[(size_t)bid * 3 + 2] = sT;
        } else {
            const int b = slice >> 6;    // slices 0..63 -> batch 0, 64..127 -> batch 1
            atomicAdd(&acc6[b * 3 + 0], sI);
            atomicAdd(&acc6[b * 3 + 1], sA);
            atomicAdd(&acc6[b * 3 + 2], sT);
        }
    }
}

__global__ void bbsd_zero(double* acc6)
{
    if (threadIdx.x < 6) acc6[threadIdx.x] = 0.0;
}

__global__ __launch_bounds__(256)
void bbsd_final(const double* __restrict__ acc6, const double* __restrict__ part,
                float* __restrict__ out, int use_part)
{
    __shared__ double sred[8][3];
    const int tid = threadIdx.x;
    if (use_part) {
        // 1024 blocks * 3 doubles; thread t sums 4 contiguous blocks (batch boundary at
        // block 512 == thread 128 == wave 4, so no wave straddles batches).
        double l0 = 0.0, l1 = 0.0, l2 = 0.0;
        for (int j = 0; j < 4; ++j) {
            const double* p = part + ((size_t)(tid * 4 + j)) * 3;
            l0 += p[0]; l1 += p[1]; l2 += p[2];
        }
        for (int off = 16; off; off >>= 1) {
            l0 += __shfl_down(l0, off);
            l1 += __shfl_down(l1, off);
            l2 += __shfl_down(l2, off);
        }
        if ((tid & 31) == 0) { sred[tid >> 5][0] = l0; sred[tid >> 5][1] = l1; sred[tid >> 5][2] = l2; }
        __syncthreads();
        if (tid == 0) {
            double I[2] = {0, 0}, A[2] = {0, 0}, T[2] = {0, 0};
            for (int w = 0; w < 8; ++w) {
                const int b = w >> 2;
                I[b] += sred[w][0]; A[b] += sred[w][1]; T[b] += sred[w][2];
            }
            double s = 0.0;
            for (int b = 0; b < 2; ++b)
                s += (T[b] == 0.0) ? 0.0 : 1.0 - 2.0 * I[b] / (A[b] + T[b] + 2e-6);
            out[0] = (float)(s * 0.5);
        }
    } else {
        if (tid == 0) {
            double s = 0.0;
            for (int b = 0; b < 2; ++b) {
                const double Ib = acc6[b * 3 + 0], Ab = acc6[b * 3 + 1], Tb = acc6[b * 3 + 2];
                s += (Tb == 0.0) ? 0.0 : 1.0 - 2.0 * Ib / (Ab + Tb + 2e-6);
            }
            out[0] = (float)(s * 0.5);
        }
    }
}

extern "C" void kernel_launch(void* const* d_in, const int* in_sizes, int n_in,
                              void* d_out, int out_size, void* d_ws, size_t ws_size,
                              hipStream_t stream)
{
    (void)in_sizes; (void)n_in; (void)out_size;
    const float* outputs = (const float*)d_in[0];
    const int*   masks   = (const int*)d_in[1];
    float*  out  = (float*)d_out;
    double* acc6 = (double*)d_ws;                       // 6 doubles (atomic fallback)
    double* part = (double*)((char*)d_ws + 64);         // 1024*3 doubles (partials)
    const int use_part = (ws_size >= 64 + (size_t)NSLICE * NBAND * 3 * sizeof(double)) ? 1 : 0;

    if (!use_part) bbsd_zero<<<1, 32, 0, stream>>>(acc6);

    dim3 grid(NBAND, NSLICE);                            // 8 bands x 128 slices = 1024 blocks
    bbsd_main<<<grid, 256, 0, stream>>>(outputs, masks, acc6, part, use_part);
    bbsd_final<<<1, 256, 0, stream>>>(acc6, part, out, use_part);
}